// LoadingModel_90838558310612
// MI455X (gfx1250) — compile-verified
//
#include <hip/hip_runtime.h>
#include <math.h>

typedef __attribute__((ext_vector_type(2))) float v2f;
typedef __attribute__((ext_vector_type(8))) float v8f;

#define P_DIM 50000
#define N_DIM 1000
#define Z_DIM 20
#define L_DIM 10
#define NBLK  ((P_DIM + 255) / 256)   // 196 blocks over p

// ---- workspace layout (float offsets) ----
#define WS_ZTX  0                       // [20 * 50000]
#define WS_W    (Z_DIM * P_DIM)         // [20 * 50000]
#define WS_RTZ  (2 * Z_DIM * P_DIM)     // [50000]
#define WS_LOG  (WS_RTZ + P_DIM)        // [50000]
#define WS_ER   (WS_LOG + P_DIM)        // [50000]
#define WS_PART (WS_ER + P_DIM)         // [NBLK] per-block maxes
#define WS_MAXV (WS_PART + 256)         // [200] global max per (k,l)
#define WS_DEN  (WS_MAXV + 256)         // [200] softmax denom per (k,l)
#define WS_ARR  (WS_DEN + 256)          // [200] uint arrival counters

// ---------------------------------------------------------------------------
// K0: reset arrival counters (must run every launch: ws persists across replays)
__global__ void k_init(unsigned* __restrict__ arrive) {
    int i = threadIdx.x;
    if (i < Z_DIM * L_DIM) arrive[i] = 0u;
}

// ---------------------------------------------------------------------------
// K1: ZtX[z,p] = sum_n mean_z[n,z] * data[n,p]  via V_WMMA_F32_16X16X4_F32.
// One wave: 16 p-columns, two 16-row z tiles. mean_z is staged in LDS padded
// to 32 z-columns (zeros for z>=20) so both A tiles load without divergence.
#define KCHUNK 200                      // rows staged per LDS chunk (mult. of 4)
__global__ __launch_bounds__(256)
void k_ztx_wmma(const float* __restrict__ mean_z,
                const float* __restrict__ data,
                float* __restrict__ ztx) {
    __shared__ float smz[KCHUNK * 32];             // 25.6 KB, zero-padded
    const int lane  = threadIdx.x & 31;
    const int wave  = threadIdx.x >> 5;
    const int ntile = blockIdx.x * 8 + wave;       // 3125 tiles of 16 columns
    const int live  = (ntile < P_DIM / 16);        // wave-uniform
    const int m     = lane & 15;
    const int kb    = (lane >> 4) * 2;             // K sub-rows 0/1 vs 2/3
    const int col   = ntile * 16 + m;

    v8f c0 = {};
    v8f c1 = {};

    for (int ch = 0; ch < N_DIM; ch += KCHUNK) {
        for (int t = threadIdx.x; t < KCHUNK * 32; t += 256) {
            const int row = t >> 5, c = t & 31;
            smz[t] = (c < Z_DIM) ? mean_z[(ch + row) * Z_DIM + c] : 0.0f;
        }
        __syncthreads();

        if (live) {
            for (int kk = 0; kk < KCHUNK; kk += 4) {
                const int rl = kk + kb;
                v2f a0, a1, b;
                a0.x = smz[rl * 32 + m];            // z rows 0..15
                a0.y = smz[(rl + 1) * 32 + m];
                a1.x = smz[rl * 32 + 16 + m];       // z rows 16..31 (padded 0)
                a1.y = smz[(rl + 1) * 32 + 16 + m];
                const long rg = (long)(ch + rl) * P_DIM + col;
                b.x = data[rg];
                b.y = data[rg + P_DIM];
                __builtin_prefetch(&data[rg + 16 * (long)P_DIM], 0, 1);

                c0 = __builtin_amdgcn_wmma_f32_16x16x4_f32(
                        false, a0, false, b, (short)0, c0, false, false);
                c1 = __builtin_amdgcn_wmma_f32_16x16x4_f32(
                        false, a1, false, b, (short)0, c1, false, false);
            }
        }
        __syncthreads();
    }

    if (!live) return;
    const int mbase = 8 * (lane >> 4);
    const int ccol  = ntile * 16 + m;
    #pragma unroll
    for (int v = 0; v < 8; ++v) {
        const int zr0 = v + mbase;
        if (zr0 < Z_DIM) ztx[(long)zr0 * P_DIM + ccol] = c0[v];
        const int zr1 = 16 + v + mbase;
        if (zr1 < Z_DIM) ztx[(long)zr1 * P_DIM + ccol] = c1[v];
    }
}

// ---------------------------------------------------------------------------
// K2: W[z,p] = sum_l mean_w[l,z,p] * alpha[l,z,p]   (float4 streams)
__global__ __launch_bounds__(256)
void k_collapse_w(const float* __restrict__ mw, const float* __restrict__ al,
                  float* __restrict__ W) {
    const int i = blockIdx.x * 256 + threadIdx.x;
    const int n4 = (Z_DIM * P_DIM) / 4;
    if (i >= n4) return;
    const float4* m4 = (const float4*)mw;
    const float4* a4 = (const float4*)al;
    float4 acc = make_float4(0.f, 0.f, 0.f, 0.f);
    #pragma unroll
    for (int l = 0; l < L_DIM; ++l) {
        float4 mv = m4[(long)l * n4 + i];
        float4 av = a4[(long)l * n4 + i];
        acc.x += mv.x * av.x; acc.y += mv.y * av.y;
        acc.z += mv.z * av.z; acc.w += mv.w * av.w;
    }
    ((float4*)W)[i] = acc;
}

// ---------------------------------------------------------------------------
// K3: var_w[l,k] = 1 / (tau * E_zzk + tau_0[l,k])   (scan-independent)
__global__ void k_varw(const float* __restrict__ mzz, const float* __restrict__ tau,
                       const float* __restrict__ tau0, float* __restrict__ varw_out) {
    const int i = blockIdx.x * blockDim.x + threadIdx.x;
    if (i >= L_DIM * Z_DIM) return;
    const int k = i % Z_DIM;
    varw_out[i] = 1.0f / (tau[0] * mzz[k * Z_DIM + k] + tau0[i]);
}

// ---------------------------------------------------------------------------
// K4: fused CAVI step.
//   Phase A (pl>=0): apply deferred finish of step (pk,pl): alpha/mean outputs
//                    and in-place W[pk] update — purely per-element.
//   Phase B (l>=0):  (l==0: compute RtZk), logits + E_RtZk, per-block max,
//                    then last-block computes global max + fixed-order
//                    sum(exp(logit-max)) -> maxv/denom for the NEXT launch.
__global__ __launch_bounds__(256)
void k_step(float* __restrict__ W, float* __restrict__ rtz,
            float* __restrict__ er, float* __restrict__ logits,
            const float* __restrict__ ztx,
            const float* __restrict__ mw_in, const float* __restrict__ al_in,
            const float* __restrict__ pi, const float* __restrict__ mzz,
            const float* __restrict__ tau_p, const float* __restrict__ tau0,
            float* __restrict__ mw_out, float* __restrict__ al_out,
            float* __restrict__ partmax, float* __restrict__ maxv_arr,
            float* __restrict__ denom_arr, unsigned* __restrict__ arrive,
            int k, int l, int pk, int pl) {
    const int i = blockIdx.x * 256 + threadIdx.x;
    const float tau = tau_p[0];

    // ---- Phase A: deferred finish of (pk, pl) ----
    if (pl >= 0 && i < P_DIM) {
        const int   ps   = pk * L_DIM + pl;
        const float mx   = maxv_arr[ps];
        const float dn   = denom_arr[ps];
        const float pEzz = mzz[pk * Z_DIM + pk];
        const float pt0  = tau0[pl * Z_DIM + pk];
        const float varn = 1.0f / (tau * pEzz + pt0);
        const long  ro   = (long)(pl * Z_DIM + pk) * P_DIM + i;
        const float a_new = expf(logits[i] - mx) / dn;
        const float m_new = tau * varn * er[i];
        const float wkl   = W[(long)pk * P_DIM + i] - mw_in[ro] * al_in[ro];
        W[(long)pk * P_DIM + i] = wkl + m_new * a_new;
        mw_out[ro] = m_new;
        al_out[ro] = a_new;
    }
    if (l < 0) return;   // final cleanup launch (uniform branch)

    // ---- Phase B: logits for (k, l) ----
    const float Ezz  = mzz[k * Z_DIM + k];
    const float t0   = tau0[l * Z_DIM + k];
    const float s2   = 1.0f / (Ezz * tau);
    const float zsc  = sqrtf(Ezz * tau) / Ezz;
    const float s0i  = 1.0f / t0;
    const float s2p  = s2 + s0i;
    const float base = 0.5f * (logf(s2) - logf(s2p));
    const float wq   = 0.5f * (s0i / s2p);

    float lval = -3.0e38f;
    if (i < P_DIM) {
        float r;
        if (l == 0) {   // RtZk with fully-updated W (phase A already applied)
            float acc = 0.f;
            #pragma unroll
            for (int j = 0; j < Z_DIM; ++j)
                acc += mzz[k * Z_DIM + j] * W[(long)j * P_DIM + i];
            r = ztx[(long)k * P_DIM + i] - (acc - Ezz * W[(long)k * P_DIM + i]);
            rtz[i] = r;
        } else {
            r = rtz[i];
        }
        const long  ro  = (long)(l * Z_DIM + k) * P_DIM + i;
        const float wkl = W[(long)k * P_DIM + i] - mw_in[ro] * al_in[ro];
        const float e   = r - Ezz * wkl;
        er[i] = e;
        const float Zs = e * zsc;
        lval = logf(pi[(long)k * P_DIM + i]) + base + Zs * Zs * wq;
        logits[i] = lval;
    }

    __shared__ float red[256];
    __shared__ int   isLast;
    red[threadIdx.x] = lval;
    __syncthreads();
    for (int s = 128; s > 0; s >>= 1) {
        if (threadIdx.x < s)
            red[threadIdx.x] = fmaxf(red[threadIdx.x], red[threadIdx.x + s]);
        __syncthreads();
    }
    const int slot = k * L_DIM + l;
    if (threadIdx.x == 0) {
        partmax[blockIdx.x] = red[0];
        __threadfence();
        unsigned old = atomicAdd(&arrive[slot], 1u);
        isLast = (old == (unsigned)(gridDim.x - 1));
    }
    __syncthreads();

    if (isLast) {                     // uniform within block
        __threadfence();
        // exact (order-independent) global max over per-block partials
        float m = -3.0e38f;
        for (int b = threadIdx.x; b < NBLK; b += 256) m = fmaxf(m, partmax[b]);
        red[threadIdx.x] = m;
        __syncthreads();
        for (int s = 128; s > 0; s >>= 1) {
            if (threadIdx.x < s)
                red[threadIdx.x] = fmaxf(red[threadIdx.x], red[threadIdx.x + s]);
            __syncthreads();
        }
        const float mx = red[0];
        __syncthreads();
        // fixed-order (deterministic) sum of exp(logit - max)
        float sacc = 0.f;
        for (int j = threadIdx.x; j < P_DIM; j += 256)
            sacc += expf(logits[j] - mx);
        red[threadIdx.x] = sacc;
        __syncthreads();
        for (int s = 128; s > 0; s >>= 1) {
            if (threadIdx.x < s) red[threadIdx.x] += red[threadIdx.x + s];
            __syncthreads();
        }
        if (threadIdx.x == 0) {
            maxv_arr[slot]  = mx;
            denom_arr[slot] = red[0];
        }
    }
}

// ---------------------------------------------------------------------------
extern "C" void kernel_launch(void* const* d_in, const int* in_sizes, int n_in,
                              void* d_out, int out_size, void* d_ws, size_t ws_size,
                              hipStream_t stream) {
    const float* data   = (const float*)d_in[0];
    const float* mean_z = (const float*)d_in[1];
    const float* mzz    = (const float*)d_in[2];
    const float* mw     = (const float*)d_in[3];
    const float* al     = (const float*)d_in[5];
    const float* tau    = (const float*)d_in[6];
    const float* tau0   = (const float*)d_in[7];
    const float* pi     = (const float*)d_in[8];

    float* out      = (float*)d_out;
    float* mw_out   = out;                                    // [10,20,50000]
    float* varw_out = out + (long)L_DIM * Z_DIM * P_DIM;      // [10,20]
    float* al_out   = varw_out + L_DIM * Z_DIM;               // [10,20,50000]

    float*    ws      = (float*)d_ws;
    float*    ztx     = ws + WS_ZTX;
    float*    W       = ws + WS_W;
    float*    rtz     = ws + WS_RTZ;
    float*    logits  = ws + WS_LOG;
    float*    er      = ws + WS_ER;
    float*    partmax = ws + WS_PART;
    float*    maxv    = ws + WS_MAXV;
    float*    denom   = ws + WS_DEN;
    unsigned* arrive  = (unsigned*)(ws + WS_ARR);

    k_init<<<1, 256, 0, stream>>>(arrive);

    k_ztx_wmma<<<(P_DIM / 16 + 7) / 8, 256, 0, stream>>>(mean_z, data, ztx);

    k_collapse_w<<<((Z_DIM * P_DIM / 4) + 255) / 256, 256, 0, stream>>>(mw, al, W);
    k_varw<<<1, 256, 0, stream>>>(mzz, tau, tau0, varw_out);

    int pk = -1, pl = -1;
    for (int k = 0; k < Z_DIM; ++k) {
        for (int l = 0; l < L_DIM; ++l) {
            k_step<<<NBLK, 256, 0, stream>>>(W, rtz, er, logits, ztx, mw, al, pi,
                                             mzz, tau, tau0, mw_out, al_out,
                                             partmax, maxv, denom, arrive,
                                             k, l, pk, pl);
            pk = k; pl = l;
        }
    }
    // final deferred finish of (19, 9); l = -1 -> phase A only
    k_step<<<NBLK, 256, 0, stream>>>(W, rtz, er, logits, ztx, mw, al, pi,
                                     mzz, tau, tau0, mw_out, al_out,
                                     partmax, maxv, denom, arrive,
                                     -1, -1, pk, pl);
}